// InfoNCE_Loss_41661182771201
// MI455X (gfx1250) — compile-verified
//
#include <hip/hip_runtime.h>
#include <hip/hip_bf16.h>
#include <stdint.h>

// InfoNCE / CPC loss for MI455X (gfx1250).
//  - fp32 WMMA (V_WMMA_F32_16X16X4_F32) for the 1x1-conv GEMM (dominant FLOPs)
//  - Tensor Data Mover (tensor_load_to_lds) double-buffers W_k chunks into LDS
//    with hardware row padding (bank-conflict-free ds_load_b64 B-fragments)
//  - threefry2x32 (JAX PRNG) computed inline for negative-sample indices
//  - deterministic two-stage reduction (no float atomics) for graph replay

typedef float v2f __attribute__((ext_vector_type(2)));
typedef float v8f __attribute__((ext_vector_type(8)));
typedef unsigned int u32x4 __attribute__((ext_vector_type(4)));
typedef int i32x4 __attribute__((ext_vector_type(4)));
typedef int i32x8 __attribute__((ext_vector_type(8)));

#define NEG_ 16
#define B_   32
#define C_   256
#define H_   32
#define W_   32
#define K_   5
#define EPS_ 1e-11f
#define LOSS_BLOCKS 240

#define CHUNK_C   64            // channels staged per TDM chunk
#define LDS_ROW   68            // 64 data floats + 4 pad floats (TDM pad feature)
#define LDS_CHUNK (C_ * LDS_ROW)  // floats per chunk buffer (256 rows)

// ---------------- threefry2x32 (Random123 / JAX default, 20 rounds) ----------
__host__ __device__ inline uint32_t rotl32(uint32_t x, int r) {
  return (x << r) | (x >> (32 - r));
}

__host__ __device__ inline void threefry2x32(uint32_t k0, uint32_t k1,
                                             uint32_t x0, uint32_t x1,
                                             uint32_t& o0, uint32_t& o1) {
  const int R[8] = {13, 15, 26, 6, 17, 29, 16, 24};
  uint32_t ks[3] = {k0, k1, 0x1BD11BDAu ^ k0 ^ k1};
  x0 += ks[0]; x1 += ks[1];
  #pragma unroll
  for (int i = 0; i < 5; ++i) {
    #pragma unroll
    for (int r = 0; r < 4; ++r) {
      x0 += x1;
      x1 = rotl32(x1, R[(i * 4 + r) & 7]);
      x1 ^= x0;
    }
    x0 += ks[(i + 1) % 3];
    x1 += ks[(i + 2) % 3] + (uint32_t)(i + 1);
  }
  o0 = x0; o1 = x1;
}

// ---------------- TDM: DMA a [256 rows x 64 cols] f32 tile of W_k to LDS -----
// Tile row stride in memory = 256 floats (W_k is [256,256] row-major).
// LDS destination gets 4 pad DWORDs after every 64 DWORDs (row stride 68 floats)
// so B-fragment ds_load_b64 across 32 lanes touches 64 distinct banks.
__device__ inline void tdm_load_wk_chunk(const float* gsrc, uint32_t lds_byte_addr) {
  const uint64_t ga = (uint64_t)(uintptr_t)gsrc;

  u32x4 g0;
  g0[0] = 1u;                                    // count=1 (valid), no gather
  g0[1] = lds_byte_addr;                         // LDS destination (bytes)
  g0[2] = (uint32_t)(ga & 0xFFFFFFFFu);          // global_addr[31:0]
  g0[3] = (uint32_t)((ga >> 32) & 0x01FFFFFFu)   // global_addr[56:32]
          | 0x80000000u;                         // type=2 ("image") in [127:126]

  i32x8 g1;
  // dword0: data_size=2 (4B) <<16 | pad_enable<<20 | pad_interval=5 (64 DW)<<22
  //         | pad_amount=3 (4 DW)<<25 ; workgroup_mask=0 (not in cluster)
  g1[0] = (int)((2u << 16) | (1u << 20) | (5u << 22) | (3u << 25));
  g1[1] = (int)((uint32_t)CHUNK_C << 16);        // tensor_dim0[15:0] @ bits 63:48
  g1[2] = (int)((uint32_t)C_ << 16);             // tensor_dim0 hi=0 | tensor_dim1 lo
  g1[3] = (int)((uint32_t)CHUNK_C << 16);        // tensor_dim1 hi=0 | tile_dim0=64
  g1[4] = (int)C_;                               // tile_dim1=256 | tile_dim2=0
  g1[5] = (int)C_;                               // tensor_dim0_stride=256 (lo)
  g1[6] = 0;                                     // stride hi | tensor_dim1_stride lo
  g1[7] = 0;

  i32x4 gz = {0, 0, 0, 0};
#if __has_include(<hip/amd_detail/amd_gfx1250_TDM.h>)
  i32x8 gz8 = {0, 0, 0, 0, 0, 0, 0, 0};
  __builtin_amdgcn_tensor_load_to_lds(g0, g1, gz, gz, gz8, 0);  // clang-23 form
#else
  __builtin_amdgcn_tensor_load_to_lds(g0, g1, gz, gz, 0);       // ROCm 7.2 form
#endif
}

// ---------------- Pass 1: GEMM  flat[n,256] = A[n,256] * Wk^T ----------------
// A[r,c] = z[b, c, h+off, w] with r = ((h*32 + w)*32 + b).
// One wave computes a 16(M) x 256(N) strip: 16 accumulator tiles. W_k is staged
// through LDS in 4 double-buffered TDM chunks of 64 channels. 8 waves/block;
// n is a multiple of 128 rows, so activity is block-uniform (barriers safe).
__global__ __launch_bounds__(256)
void gemm_ztwk_kernel(const float* __restrict__ z, const float* __restrict__ Wfull,
                      const int* __restrict__ skip_p, float* __restrict__ out, int k) {
  __shared__ float wlds[2 * LDS_CHUNK];          // 2 x ~68KB double buffer

  const int skip = skip_p[0];
  const int off  = k + skip;
  const int Hp   = H_ - off;
  const int n    = Hp * W_ * B_;                 // multiple of 1024

  const int lane  = threadIdx.x & 31;
  const int wave  = threadIdx.x >> 5;
  const int mtile = blockIdx.x * 8 + wave;
  if (blockIdx.x * 128 >= n) return;             // whole block inactive

  const float* Wk = Wfull + (size_t)(k - 1) * C_ * C_;
  const uint32_t lds_base = (uint32_t)(uintptr_t)(void*)&wlds[0];

  // A fragment addressing (16x4 f32 layout: lanes 0-15 K=0,1; lanes 16-31 K=2,3)
  const int r = mtile * 16 + (lane & 15);
  const int b = r & 31;
  const int w = (r >> 5) & 31;
  const int h = r >> 10;
  const float* zrow = z + (size_t)b * (C_ * H_ * W_) + (size_t)(h + off) * W_ + w;
  const int klo   = (lane >> 4) * 2;
  const int obase = lane & 15;

  v8f zero = {0.f, 0.f, 0.f, 0.f, 0.f, 0.f, 0.f, 0.f};
  v8f acc[16];
  #pragma unroll
  for (int ot = 0; ot < 16; ++ot) acc[ot] = zero;

  const int NCHUNK = C_ / CHUNK_C;               // 4
  if (wave == 0)                                 // prologue: chunk 0 -> buffer 0
    tdm_load_wk_chunk(Wk, lds_base);

  for (int ch = 0; ch < NCHUNK; ++ch) {
    if (wave == 0) {
      if (ch + 1 < NCHUNK) {                     // async: next chunk -> other buf
        tdm_load_wk_chunk(Wk + (ch + 1) * CHUNK_C,
                          lds_base + (uint32_t)(((ch + 1) & 1) * LDS_CHUNK * 4));
        // TDM completes in-order per wave: cnt<=1 means chunk `ch` is resident
        __builtin_amdgcn_s_wait_tensorcnt(1);
      } else {
        __builtin_amdgcn_s_wait_tensorcnt(0);    // drain: last chunk resident
      }
    }
    __syncthreads();                             // publish buffer to all 8 waves

    const float* buf = wlds + (size_t)(ch & 1) * LDS_CHUNK;
    const float* zch = zrow + (size_t)(ch * CHUNK_C) * (H_ * W_);

    for (int c0 = 0; c0 < CHUNK_C; c0 += 4) {
      v2f a;
      a.x = zch[(size_t)(c0 + klo) * (H_ * W_)];
      a.y = zch[(size_t)(c0 + klo + 1) * (H_ * W_)];
      if (c0 + 8 < CHUNK_C)                      // keep next z K-step warm
        __builtin_prefetch(zch + (size_t)(c0 + 8 + klo) * (H_ * W_), 0, 3);

      // Phase 1: issue all 16 B-fragment ds_load_b64 into distinct registers
      // so DScnt waits retire incrementally instead of serializing each WMMA.
      v2f bf[16];
      #pragma unroll
      for (int ot = 0; ot < 16; ++ot) {
        const float* bp = buf + (size_t)(ot * 16 + obase) * LDS_ROW + (c0 + klo);
        bf[ot] = *(const v2f*)bp;
      }
      // Phase 2: 16 back-to-back WMMAs overlapping remaining LDS latency.
      #pragma unroll
      for (int ot = 0; ot < 16; ++ot) {
        acc[ot] = __builtin_amdgcn_wmma_f32_16x16x4_f32(
            /*neg_a=*/false, a, /*neg_b=*/false, bf[ot],
            /*c_mod=*/(short)0, acc[ot], /*reuse_a=*/false, /*reuse_b=*/false);
      }
    }
    __syncthreads();                             // all waves done before reuse
  }

  // D layout: lane&15 = N column, VGPR v = row M=v (lanes<16) / v+8 (lanes>=16)
  const int mhi = (lane >> 4) * 8;
  #pragma unroll
  for (int ot = 0; ot < 16; ++ot) {
    #pragma unroll
    for (int v = 0; v < 8; ++v) {
      const int mrow = mtile * 16 + mhi + v;
      out[(size_t)mrow * C_ + ot * 16 + obase] = acc[ot][v];
    }
  }
}

// ---------------- Pass 2: logits + softmax NLL, per-block partial sums -------
__global__ __launch_bounds__(256)
void loss_kernel(const float* __restrict__ cbuf, const float* __restrict__ flat,
                 const int* __restrict__ skip_p, float* __restrict__ partials,
                 uint32_t key0, uint32_t key1, int k) {
  const int skip = skip_p[0];
  const int off  = k + skip;
  const int Hp   = H_ - off;
  const int n    = Hp * W_ * B_;
  const uint32_t half = (uint32_t)n * NEG_ / 2u;   // threefry counter split point

  const int lane   = threadIdx.x & 31;
  const int wave   = threadIdx.x >> 5;
  const int gwave  = blockIdx.x * 8 + wave;
  const int nwaves = gridDim.x * 8;

  __shared__ float wsum[8];
  float acc = 0.f;

  for (int r = gwave; r < n; r += nwaves) {
    const int b = r & 31;
    const int w = (r >> 5) & 31;
    const int h = r >> 10;
    const float* cp = cbuf + (size_t)b * (C_ * H_ * W_) + (size_t)h * W_ + w;

    float ctx[8];
    #pragma unroll
    for (int t = 0; t < 8; ++t)
      ctx[t] = cp[(size_t)(lane + 32 * t) * (H_ * W_)];

    float logits[1 + NEG_];

    {  // positive logit
      const float* fr = flat + (size_t)r * C_;
      float s = 0.f;
      #pragma unroll
      for (int t = 0; t < 8; ++t) s += ctx[t] * fr[lane + 32 * t];
      #pragma unroll
      for (int d = 16; d > 0; d >>= 1) s += __shfl_xor(s, d, 32);
      logits[0] = s;
    }

    #pragma unroll
    for (int j = 0; j < NEG_; ++j) {  // negatives via threefry counter mode
      const uint32_t i = (uint32_t)r * NEG_ + (uint32_t)j;
      const uint32_t p = (i < half) ? i : (i - half);
      uint32_t b0, b1;
      threefry2x32(key0, key1, p, p + half, b0, b1);
      const uint32_t bits = (i < half) ? b0 : b1;
      const int row = (int)(bits % (uint32_t)n);
      const float* fr = flat + (size_t)row * C_;
      float s = 0.f;
      #pragma unroll
      for (int t = 0; t < 8; ++t) s += ctx[t] * fr[lane + 32 * t];
      #pragma unroll
      for (int d = 16; d > 0; d >>= 1) s += __shfl_xor(s, d, 32);
      logits[1 + j] = s;
    }

    // softmax over 17 candidates, -log(p0 + eps)
    float m = logits[0];
    #pragma unroll
    for (int j = 1; j < 1 + NEG_; ++j) m = fmaxf(m, logits[j]);
    float se = 0.f;
    #pragma unroll
    for (int j = 0; j < 1 + NEG_; ++j) se += expf(logits[j] - m);
    const float p0 = expf(logits[0] - m) / se;
    if (lane == 0) acc += -logf(p0 + EPS_);
  }

  if (lane == 0) wsum[wave] = acc;
  __syncthreads();
  if (threadIdx.x == 0) {
    float s = 0.f;
    for (int i = 0; i < 8; ++i) s += wsum[i];
    partials[(size_t)(k - 1) * gridDim.x + blockIdx.x] = s;
  }
}

// ---------------- finalize: fixed-order sum, overwrite d_out -----------------
__global__ void finalize_kernel(const float* __restrict__ partials,
                                const int* __restrict__ skip_p,
                                float* __restrict__ out, int G) {
  if (threadIdx.x == 0 && blockIdx.x == 0) {
    const int skip = skip_p[0];
    float total = 0.f;
    for (int k = 1; k <= K_; ++k) {
      const int n = (H_ - (k + skip)) * W_ * B_;
      float s = 0.f;
      for (int i = 0; i < G; ++i) s += partials[(k - 1) * G + i];
      total += s / (float)n;
    }
    out[0] = total / (float)K_;
  }
}

extern "C" void kernel_launch(void* const* d_in, const int* in_sizes, int n_in,
                              void* d_out, int out_size, void* d_ws, size_t ws_size,
                              hipStream_t stream) {
  const float* z    = (const float*)d_in[0];
  const float* cc   = (const float*)d_in[1];
  const float* Wt   = (const float*)d_in[2];
  const int*   skip = (const int*)d_in[3];
  float* out = (float*)d_out;

  // ws layout: [0, 30720*256) floats = ztwk scratch (max over k, skip=1);
  // then K_*LOSS_BLOCKS floats of per-block partial sums.
  float* ztwk = (float*)d_ws;
  float* partials = (float*)d_ws + (size_t)30720 * 256;
  const int G = LOSS_BLOCKS;

  for (int k = 1; k <= K_; ++k) {
    // fold_in(jax.random.key(1), k): master threefry key = (0,1), data = (0,k)
    uint32_t key0, key1;
    threefry2x32(0u, 1u, 0u, (uint32_t)k, key0, key1);

    const int Hp_assumed = 31 - k;             // grid sized for skip_step == 1
    const int gemm_blocks = Hp_assumed * 8;    // (Hp*1024)/16 tiles / 8 waves
    gemm_ztwk_kernel<<<gemm_blocks, 256, 0, stream>>>(z, Wt, skip, ztwk, k);
    loss_kernel<<<G, 256, 0, stream>>>(cc, ztwk, skip, partials, key0, key1, k);
  }
  finalize_kernel<<<1, 32, 0, stream>>>(partials, skip, out, G);
}